// VanillaMLA_63436666962255
// MI455X (gfx1250) — compile-verified
//
#include <hip/hip_runtime.h>
#include <hip/hip_bf16.h>

typedef __attribute__((ext_vector_type(2))) float v2f;
typedef __attribute__((ext_vector_type(8))) float v8f;

#define B_     8
#define QLEN_  4
#define KVLEN_ 4096
#define HID_   2048
#define H_     16
#define QLORA_ 1536
#define KVLORA_ 512
#define NOPE_  128
#define ROPE_  64
#define VHD_   128
#define QHD_   192
#define M_ROWS (B_*QLEN_)          // 32

#define WMMA_F32(acc, a, b) \
    __builtin_amdgcn_wmma_f32_16x16x4_f32(false, (a), false, (b), (short)0, (acc), false, false)

// -------------------------------------------------------------------------
// Generic fp32 WMMA GEMM:  C[m,n] = sum_k A[m*lda+k] * Bm[n*ldb_n + k*LDBK]
// LDBK is compile-time so B addressing folds into immediate offsets.
// One wave -> one 16x16 C tile; 4 waves/block along N; blockIdx.z batches.
// Requires: M,N multiples of 16; K multiple of 16.
// Inner loop processes 16 K-values: batch all loads, then 4 back-to-back WMMAs.
// -------------------------------------------------------------------------
template<int LDBK>
__global__ __launch_bounds__(128) void gemm_wmma_f32(
    const float* __restrict__ A, int lda, long long sAz,
    const float* __restrict__ Bm, int ldb_n, long long sBz,
    float* __restrict__ C, int ldc, long long sCz,
    int M, int N, int K)
{
    const int lane  = threadIdx.x & 31;
    const int wave  = threadIdx.x >> 5;
    const int tileN = (blockIdx.x * 4 + wave) * 16;
    const int tileM = blockIdx.y * 16;
    if (tileN >= N || tileM >= M) return;   // wave-uniform

    A  += (long long)blockIdx.z * sAz;
    Bm += (long long)blockIdx.z * sBz;
    C  += (long long)blockIdx.z * sCz;

    const int mrow = lane & 15;            // M index (A) / N index (B)
    const int kh   = (lane >> 4) << 1;     // 0 for lanes 0-15, 2 for 16-31

    const float* Arow = A  + (size_t)(tileM + mrow) * lda;
    const float* Brow = Bm + (size_t)(tileN + mrow) * ldb_n;

    v8f acc = {};
    for (int k0 = 0; k0 < K; k0 += 16) {
        v2f a[4], b[4];
        #pragma unroll
        for (int i = 0; i < 4; ++i)
            a[i] = *(const v2f*)&Arow[k0 + 4 * i + kh];
        #pragma unroll
        for (int i = 0; i < 4; ++i) {
            if (LDBK == 1) {
                b[i] = *(const v2f*)&Brow[k0 + 4 * i + kh];
            } else {
                b[i].x = Brow[(size_t)(k0 + 4 * i + kh)     * LDBK];
                b[i].y = Brow[(size_t)(k0 + 4 * i + kh + 1) * LDBK];
            }
        }
        #pragma unroll
        for (int i = 0; i < 4; ++i)
            acc = WMMA_F32(acc, a[i], b[i]);
    }

    const int rbase = (lane < 16) ? 0 : 8;
    const int col   = lane & 15;
    #pragma unroll
    for (int v = 0; v < 8; ++v)
        C[(size_t)(tileM + rbase + v) * ldc + tileN + col] = acc[v];
}

// -------------------------------------------------------------------------
// In-place RMSNorm over rows of x [nrows x ncols], weight w.
// -------------------------------------------------------------------------
__global__ __launch_bounds__(256) void rmsnorm_rows(
    float* __restrict__ x, const float* __restrict__ w, int ncols)
{
    float* xr = x + (size_t)blockIdx.x * ncols;
    __shared__ float red[256];
    float s = 0.f;
    for (int i = threadIdx.x; i < ncols; i += 256) { float v = xr[i]; s += v * v; }
    red[threadIdx.x] = s;
    __syncthreads();
    for (int off = 128; off > 0; off >>= 1) {
        if (threadIdx.x < off) red[threadIdx.x] += red[threadIdx.x + off];
        __syncthreads();
    }
    float rs = rsqrtf(red[0] / (float)ncols + 1e-6f);
    for (int i = threadIdx.x; i < ncols; i += 256) xr[i] = xr[i] * rs * w[i];
}

// -------------------------------------------------------------------------
// Absorbed-MLA attention core.  One block (256 threads = 8 wave32) per (b,h).
//   scores[q,k] = (q_abs*ss)·c_kv[k] + (q_pe*rs)·k_pe[k]   (WMMA, K=576)
//   softmax over k=4096 in LDS
//   o_lat[q,c]  = attn[q,:] @ c_kv[:,c]                    (WMMA, K=4096)
// Dynamic LDS: qs 16x576 | scores 4x4096 | red 256  (= 103,424 B, <320KB WGP)
// -------------------------------------------------------------------------
#define ATTN_LDS_FLOATS (16*576 + 4*KVLEN_ + 256)

__global__ __launch_bounds__(256) void attn_kernel(
    const float* __restrict__ q_abs,   // [B,Q,H,512]
    const float* __restrict__ qbuf,    // [B*Q, H*192]
    const float* __restrict__ c_kv,    // [B,KVLEN,512]
    const float* __restrict__ k_pe,    // [B,KVLEN,64]
    float* __restrict__ o_lat)         // [B,Q,H,512]
{
    extern __shared__ float smem[];
    float* qs       = smem;                  // [16][576], rows 4-15 zero
    float* scores_s = smem + 16 * 576;       // [4][4096]
    float* sred     = scores_s + 4 * KVLEN_; // [256]

    const int h = blockIdx.x;
    const int b = blockIdx.y;
    const int tid  = threadIdx.x;
    const int lane = tid & 31;
    const int wave = tid >> 5;
    const int mrow = lane & 15;
    const int kh   = (lane >> 4) << 1;

    const float ss = 0.07216878364870323f;   // QHD^-0.5 = 192^-0.5
    const float rs = 0.125f;                 // ROPE^-0.5

    // ---- stage scaled, zero-padded Q tile: [q_abs*ss | q_pe*rs] ----
    for (int idx = tid; idx < 16 * 576; idx += 256) {
        int m = idx / 576, cc = idx - m * 576;
        float v = 0.f;
        if (m < QLEN_) {
            if (cc < KVLORA_)
                v = q_abs[(((size_t)b * QLEN_ + m) * H_ + h) * KVLORA_ + cc] * ss;
            else
                v = qbuf[((size_t)b * QLEN_ + m) * (H_ * QHD_) + h * QHD_ + NOPE_ + (cc - KVLORA_)] * rs;
        }
        qs[idx] = v;
    }
    __syncthreads();

    // ---- scores: 256 K-tiles of 16 keys; contraction over 576 latent dims ----
    const float* ckv_b = c_kv + (size_t)b * KVLEN_ * KVLORA_;
    const float* kpe_b = k_pe + (size_t)b * KVLEN_ * ROPE_;

    for (int t = wave; t < KVLEN_ / 16; t += 8) {
        const int key = t * 16 + mrow;
        const float* ckv_row = ckv_b + (size_t)key * KVLORA_;
        const float* kpe_row = kpe_b + (size_t)key * ROPE_;
        v8f acc = {};
        for (int c0 = 0; c0 < KVLORA_ + ROPE_; c0 += 16) {    // 512-boundary is chunk-aligned
            const float* brow = (c0 < KVLORA_) ? (ckv_row + c0) : (kpe_row + (c0 - KVLORA_));
            if (c0 + 16 < KVLORA_)
                __builtin_prefetch(&ckv_row[c0 + 16 + kh], 0, 1);
            v2f a[4], bv[4];
            #pragma unroll
            for (int i = 0; i < 4; ++i) {
                a[i]  = *(const v2f*)&qs[mrow * 576 + c0 + 4 * i + kh];
                bv[i] = *(const v2f*)&brow[4 * i + kh];
            }
            #pragma unroll
            for (int i = 0; i < 4; ++i)
                acc = WMMA_F32(acc, a[i], bv[i]);
        }
        if (lane < 16) {
            #pragma unroll
            for (int v = 0; v < QLEN_; ++v)
                scores_s[v * KVLEN_ + t * 16 + lane] = acc[v];
        }
    }
    __syncthreads();

    // ---- softmax over k (rows 0..3, 64 threads per row) ----
    {
        const int r = tid >> 6, c = tid & 63;
        float* srow = scores_s + r * KVLEN_;
        float mx = -3.0e38f;
        for (int i = c; i < KVLEN_; i += 64) mx = fmaxf(mx, srow[i]);
        sred[tid] = mx;
        __syncthreads();
        for (int off = 32; off > 0; off >>= 1) {
            if (c < off) sred[tid] = fmaxf(sred[tid], sred[tid + off]);
            __syncthreads();
        }
        float rowmax = sred[r << 6];
        float sum = 0.f;
        for (int i = c; i < KVLEN_; i += 64) {
            float e = __expf(srow[i] - rowmax);
            srow[i] = e;
            sum += e;
        }
        __syncthreads();
        sred[tid] = sum;
        __syncthreads();
        for (int off = 32; off > 0; off >>= 1) {
            if (c < off) sred[tid] += sred[tid + off];
            __syncthreads();
        }
        float inv = 1.f / sred[r << 6];
        for (int i = c; i < KVLEN_; i += 64) srow[i] *= inv;
    }
    __syncthreads();

    // ---- o_lat = attn @ c_kv : 32 N-tiles of 16 latent cols, K=4096 ----
    // A rows 4..15 alias row (mrow&3) and are scaled by 0 -> uniform LDS loads.
    {
        const float  amask = (mrow < QLEN_) ? 1.f : 0.f;
        const float* srow  = scores_s + (size_t)(mrow & 3) * KVLEN_;
        for (int nt = wave; nt < KVLORA_ / 16; nt += 8) {
            const int ccol = nt * 16 + mrow;
            const float* bcol = ckv_b + ccol;
            v8f acc = {};
            for (int k0 = 0; k0 < KVLEN_; k0 += 16) {
                if (k0 + 16 < KVLEN_)
                    __builtin_prefetch(&bcol[(size_t)(k0 + 16 + kh) * KVLORA_], 0, 1);
                v2f a[4], bv[4];
                #pragma unroll
                for (int i = 0; i < 4; ++i) {
                    const int kk = k0 + 4 * i + kh;
                    a[i] = (*(const v2f*)&srow[kk]) * amask;
                    bv[i].x = bcol[(size_t)kk       * KVLORA_];
                    bv[i].y = bcol[(size_t)(kk + 1) * KVLORA_];
                }
                #pragma unroll
                for (int i = 0; i < 4; ++i)
                    acc = WMMA_F32(acc, a[i], bv[i]);
            }
            if (lane < 16) {
                #pragma unroll
                for (int v = 0; v < QLEN_; ++v)
                    o_lat[(((size_t)b * QLEN_ + v) * H_ + h) * KVLORA_ + nt * 16 + lane] = acc[v];
            }
        }
    }
}

// -------------------------------------------------------------------------
extern "C" void kernel_launch(void* const* d_in, const int* in_sizes, int n_in,
                              void* d_out, int out_size, void* d_ws, size_t ws_size,
                              hipStream_t stream)
{
    const float* hidden = (const float*)d_in[0];  // [B,QLEN,HID]
    const float* c_kv   = (const float*)d_in[1];  // [B,KVLEN,KV_LORA]
    const float* k_pe   = (const float*)d_in[2];  // [B,KVLEN,ROPE]
    const float* Wqa    = (const float*)d_in[3];  // [Q_LORA,HID]
    const float* qa_w   = (const float*)d_in[4];  // [Q_LORA]
    const float* Wqb    = (const float*)d_in[5];  // [H*QHD, Q_LORA]
    const float* Wkvb   = (const float*)d_in[6];  // [H*(NOPE+VHD), KV_LORA]
    const float* Wo     = (const float*)d_in[7];  // [HID, H*VHD]
    float* out = (float*)d_out;

    float* ws    = (float*)d_ws;
    float* cQ    = ws;                         // 32*1536
    float* qbuf  = cQ    + M_ROWS * QLORA_;    // 32*3072
    float* q_abs = qbuf  + M_ROWS * H_ * QHD_; // 32*16*512
    float* o_lat = q_abs + M_ROWS * H_ * KVLORA_;
    float* out_v = o_lat + M_ROWS * H_ * KVLORA_; // 32*2048

    const dim3 blk(128);
    const long long HS = (long long)(NOPE_ + VHD_) * KVLORA_; // per-head Wkvb stride

    // 1) cQ = hidden @ Wqa^T ; rmsnorm
    gemm_wmma_f32<1><<<dim3(QLORA_/64, 2, 1), blk, 0, stream>>>(
        hidden, HID_, 0, Wqa, HID_, 0, cQ, QLORA_, 0, M_ROWS, QLORA_, HID_);
    rmsnorm_rows<<<M_ROWS, 256, 0, stream>>>(cQ, qa_w, QLORA_);

    // 2) q = cQ @ Wqb^T
    gemm_wmma_f32<1><<<dim3((H_*QHD_)/64, 2, 1), blk, 0, stream>>>(
        cQ, QLORA_, 0, Wqb, QLORA_, 0, qbuf, H_*QHD_, 0, M_ROWS, H_*QHD_, QLORA_);

    // 3) q_abs[.,h,c] = q_nope[.,h,:] @ Wkvb_nope[h]   (batched over heads)
    gemm_wmma_f32<KVLORA_><<<dim3(KVLORA_/64, 2, H_), blk, 0, stream>>>(
        qbuf, H_*QHD_, QHD_, Wkvb, 1, HS,
        q_abs, H_*KVLORA_, KVLORA_, M_ROWS, KVLORA_, NOPE_);

    // 4) attention in latent space
    attn_kernel<<<dim3(H_, B_), 256, ATTN_LDS_FLOATS * sizeof(float), stream>>>(
        q_abs, qbuf, c_kv, k_pe, o_lat);

    // 5) out_v[.,h,d] = o_lat[.,h,:] @ Wkvb_v[h]^T   (batched over heads)
    gemm_wmma_f32<1><<<dim3(VHD_/64, 2, H_), blk, 0, stream>>>(
        o_lat, H_*KVLORA_, KVLORA_, Wkvb + (size_t)NOPE_*KVLORA_, KVLORA_, HS,
        out_v, H_*VHD_, VHD_, M_ROWS, VHD_, KVLORA_);

    // 6) final = out_v @ Wo^T
    gemm_wmma_f32<1><<<dim3(HID_/64, 2, 1), blk, 0, stream>>>(
        out_v, H_*VHD_, 0, Wo, H_*VHD_, 0, out, HID_, 0, M_ROWS, HID_, H_*VHD_);
}